// ProposalTargetLayer_7550552507196
// MI455X (gfx1250) — compile-verified
//
#include <hip/hip_runtime.h>
#include <hip/hip_bf16.h>

// ------------------------- problem constants -------------------------
#define BB 4
#define MM 512
#define NN 32
#define PP 16384
#define CCH 128
#define ROI 64
#define FG_PER 16
#define BG_PER 48
#define NUM_PTS 512
#define NEGV (-1.0e30f)
#define KILLV (-3.0e38f)
#define REG_FG 0.55f
#define CLS_FG 0.60f
#define CLS_BG 0.45f
#define FG_TH  0.55f
#define EWID   1.0f
#define TWO_PI 6.283185307179586f

// output-region offsets (float elements) in d_out, reference return order
#define O_SAMPLED 0u
#define O_FEAT    393216u
#define O_CLS     17432576u
#define O_REG     17432832u
#define O_GT      17433088u
#define O_IOU     17435136u
#define O_ROIS    17435392u

typedef __attribute__((ext_vector_type(2))) float v2f;
typedef __attribute__((ext_vector_type(8))) float v8f;
typedef int v4i_vs __attribute__((vector_size(16)));   // matches builtin "V4i"

// ---------------- CDNA5 feature gates (device pass only) --------------
#if defined(__AMDGCN__) && __has_builtin(__builtin_amdgcn_wmma_f32_16x16x4_f32)
#define USE_WMMA_F32 1
#else
#define USE_WMMA_F32 0
#endif

#if defined(__AMDGCN__) && __has_builtin(__builtin_amdgcn_global_load_async_to_lds_b128)
#define USE_ASYNC_LDS 1
#else
#define USE_ASYNC_LDS 0
#endif

__device__ __forceinline__ unsigned ballot32(bool p) {
#if defined(__AMDGCN__) && __has_builtin(__builtin_amdgcn_ballot_w32)
  return __builtin_amdgcn_ballot_w32(p);
#else
  return (unsigned)__ballot(p);
#endif
}

__device__ __forceinline__ void wait_async0() {
#if defined(__AMDGCN__) && __has_builtin(__builtin_amdgcn_s_wait_asynccnt)
  __builtin_amdgcn_s_wait_asynccnt(0);
#elif defined(__AMDGCN__)
  asm volatile("s_wait_asynccnt 0" ::: "memory");
#endif
}

// ---------------------------------------------------------------------
// Kernel A: pairwise 3D IoU (512x32 per batch), max + argmax over GT
// grid = B, block = 512 (one thread per ROI)
// ---------------------------------------------------------------------
__global__ __launch_bounds__(512) void k_iou(const float* __restrict__ roi,
                                             const float* __restrict__ gt,
                                             float* __restrict__ max_ov,
                                             int* __restrict__ ga) {
  int b = blockIdx.x, m = threadIdx.x;
  __shared__ float g[NN * 8];
  if (m < NN * 8) g[m] = gt[b * NN * 8 + m];
  __syncthreads();

  const float* r = roi + (size_t)(b * MM + m) * 7;
  float x1 = r[0], y1 = r[1], z1 = r[2], h1 = r[3], w1 = r[4], l1 = r[5];
  float v1 = h1 * w1 * l1;
  float best = -3.4e38f;
  int bi = 0;
#pragma unroll 4
  for (int n = 0; n < NN; ++n) {
    const float* q = g + n * 8;
    float x2 = q[0], y2 = q[1], z2 = q[2], h2 = q[3], w2 = q[4], l2 = q[5];
    float ox = fminf(x1 + l1 * 0.5f, x2 + l2 * 0.5f) - fmaxf(x1 - l1 * 0.5f, x2 - l2 * 0.5f);
    float oz = fminf(z1 + w1 * 0.5f, z2 + w2 * 0.5f) - fmaxf(z1 - w1 * 0.5f, z2 - w2 * 0.5f);
    float oy = fminf(y1, y2) - fmaxf(y1 - h1, y2 - h2);
    ox = fmaxf(ox, 0.f); oy = fmaxf(oy, 0.f); oz = fmaxf(oz, 0.f);
    float inter = ox * oy * oz;
    float v2 = h2 * w2 * l2;
    float iou = inter / (v1 + v2 - inter + 1e-7f);
    if (iou > best) { best = iou; bi = n; }   // strict > keeps lowest index on tie
  }
  max_ov[b * MM + m] = best;
  ga[b * MM + m] = bi;
}

// ---------------------------------------------------------------------
// Kernel B: top-16 FG + top-48 BG per batch (tie -> lower index, as
// jax.lax.top_k). grid = B, block = 512.
// ---------------------------------------------------------------------
__global__ __launch_bounds__(512) void k_topk(const float* __restrict__ max_ov,
                                              int* __restrict__ inds) {
  int b = blockIdx.x, t = threadIdx.x;
  __shared__ float ms[MM];
  __shared__ float sv[MM];
  __shared__ int   si[MM];
  float mo = max_ov[b * MM + t];

  for (int phase = 0; phase < 2; ++phase) {
    int K   = phase == 0 ? FG_PER : BG_PER;
    int off = phase == 0 ? 0 : FG_PER;
    ms[t] = phase == 0 ? ((mo >= FG_TH) ? mo : NEGV)
                       : ((mo < CLS_BG) ? mo : NEGV);
    __syncthreads();
    for (int k = 0; k < K; ++k) {
      sv[t] = ms[t]; si[t] = t;
      __syncthreads();
      for (int s = 256; s > 0; s >>= 1) {
        if (t < s) {
          float v2 = sv[t + s]; int i2 = si[t + s];
          if (v2 > sv[t] || (v2 == sv[t] && i2 < si[t])) { sv[t] = v2; si[t] = i2; }
        }
        __syncthreads();
      }
      if (t == 0) {
        inds[b * ROI + off + k] = si[0];
        ms[si[0]] = KILLV;            // below NEGV: never re-picked
      }
      __syncthreads();
    }
  }
}

// ---------------------------------------------------------------------
// Kernel C: point-in-rotated-box + ordered compaction of first 512
// inside indices.  wave32 ballot + cross-wave LDS scan.
// grid = B*ROI, block = 512 (16 waves).
// ---------------------------------------------------------------------
__global__ __launch_bounds__(512) void k_inside(const float* __restrict__ roi,
                                                const float* __restrict__ xyz,
                                                const int* __restrict__ inds,
                                                int* __restrict__ idxp,
                                                int* __restrict__ cnt) {
  int rid = blockIdx.x;
  int b = rid >> 6;
  int t = threadIdx.x;
  int lane = t & 31, w = t >> 5;
  __shared__ float sroi[7];
  __shared__ int wsum[16];
  __shared__ int wpre[16];
  __shared__ int sbase;

  int sel = inds[rid];
  if (t < 7) sroi[t] = roi[(size_t)(b * MM + sel) * 7 + t];
  if (t == 0) sbase = 0;
  __syncthreads();

  float cx = sroi[0], cy = sroi[1], cz = sroi[2];
  float he = sroi[3] + 2.f * EWID, we = sroi[4] + 2.f * EWID, le = sroi[5] + 2.f * EWID;
  float ry = sroi[6];
  float cc = cosf(ry), ss = sinf(ry);

  for (int chunk = 0; chunk < PP / 512; ++chunk) {
    int p = chunk * 512 + t;
    const float* q = xyz + (size_t)(b * PP + p) * 3;
    float dx = q[0] - cx;
    float dy = q[1] - (cy + EWID);
    float dz = q[2] - cz;
    float lx = cc * dx + ss * dz;
    float lz = cc * dz - ss * dx;
    bool inside = (fabsf(lx) < le * 0.5f) && (fabsf(lz) < we * 0.5f) &&
                  (dy > -he) && (dy < 0.0f);
    unsigned mask = ballot32(inside);
    int rank = __popc(mask & ((1u << lane) - 1u));
    if (lane == 0) wsum[w] = __popc(mask);
    __syncthreads();
    if (t == 0) {
      int acc = sbase;
      for (int i = 0; i < 16; ++i) { wpre[i] = acc; acc += wsum[i]; }
      sbase = acc;
    }
    __syncthreads();
    if (inside) {
      int slot = wpre[w] + rank;
      if (slot < NUM_PTS) idxp[(size_t)rid * NUM_PTS + slot] = p;
    }
    __syncthreads();
    if (sbase >= NUM_PTS) break;     // uniform (shared, post-barrier)
  }

  __syncthreads();
  int cv = sbase; if (cv > NUM_PTS) cv = NUM_PTS;
  if (t == 0) cnt[rid] = cv;
  int idx0 = (cv > 0) ? idxp[(size_t)rid * NUM_PTS] : 0;
  if (t >= cv) idxp[(size_t)rid * NUM_PTS + t] = idx0;  // pad with first index
}

// ---------------------------------------------------------------------
// Kernel D: labels, gt_out, rois, roi_iou.  grid = 1, block = 256.
// ---------------------------------------------------------------------
__global__ __launch_bounds__(256) void k_labels(const float* __restrict__ roi,
                                                const float* __restrict__ gt,
                                                const float* __restrict__ max_ov,
                                                const int* __restrict__ ga,
                                                const int* __restrict__ inds,
                                                const int* __restrict__ cnt,
                                                float* __restrict__ out) {
  int t = threadIdx.x;            // 0..255 : selected ROI id
  int b = t >> 6;
  int i = inds[t];
  float iou = max_ov[b * MM + i];
  int g = ga[b * MM + i];
  const float* r  = roi + (size_t)(b * MM + i) * 7;
  const float* gb = gt + (size_t)(b * NN + g) * 8;

  float rv[7];
#pragma unroll
  for (int j = 0; j < 7; ++j) rv[j] = r[j];
#pragma unroll
  for (int j = 0; j < 7; ++j) out[O_ROIS + t * 7 + j] = rv[j];
  out[O_IOU + t] = iou;

  float rmod = fmodf(rv[6], TWO_PI);
  if (rmod < 0.f) rmod += TWO_PI;                 // python-style mod
  float c = cosf(rmod), s = sinf(rmod);
  float gx0 = gb[0] - rv[0], gy0 = gb[1] - rv[1], gz0 = gb[2] - rv[2];
  float gx = c * gx0 + s * gz0;
  float gz = c * gz0 - s * gx0;
  out[O_GT + t * 8 + 0] = gx;
  out[O_GT + t * 8 + 1] = gy0;
  out[O_GT + t * 8 + 2] = gz;
  out[O_GT + t * 8 + 3] = gb[3];
  out[O_GT + t * 8 + 4] = gb[4];
  out[O_GT + t * 8 + 5] = gb[5];
  out[O_GT + t * 8 + 6] = gb[6] - rmod;
  out[O_GT + t * 8 + 7] = gb[7];

  bool valid = cnt[t] > 0;
  int reg = (iou > REG_FG && valid) ? 1 : 0;
  int cls = (iou > CLS_FG) ? 1 : 0;
  bool amb = (iou > CLS_BG) && (iou < CLS_FG);
  if (!valid || amb) cls = -1;
  if (cls == 1) cls = (int)gb[7];
  out[O_CLS + t] = (float)cls;
  out[O_REG + t] = (float)reg;
}

// ---------------------------------------------------------------------
// Kernel E: pool_feat gather. One ROI per block, 256 threads (8 waves).
// Phase 1: each wave gathers one 128-float feature row as 32 aligned
//          float4s via GLOBAL_LOAD_ASYNC_TO_LDS_B128 (+extras to LDS).
// Phase 2: block streams 8*130 contiguous floats to HBM (coalesced).
// ---------------------------------------------------------------------
__global__ __launch_bounds__(256) void k_pool(const float* __restrict__ feat,
                                              const float* __restrict__ seg,
                                              const float* __restrict__ dep,
                                              const int* __restrict__ idxp,
                                              const int* __restrict__ cnt,
                                              float* __restrict__ out) {
  int rid = blockIdx.x;
  int b = rid >> 6;
  int t = threadIdx.x;
  int lane = t & 31, w = t >> 5;
  bool empty = (cnt[rid] == 0);
  float* outf = out + O_FEAT + (size_t)rid * NUM_PTS * 130;

  __shared__ float lds[8 * 132];    // 132-stride: features 16B-aligned, extras at +128

  for (int grp = 0; grp < NUM_PTS / 8; ++grp) {
    if (!empty) {
      int row = grp * 8 + w;
      int p = idxp[(size_t)rid * NUM_PTS + row];
      const float* src = feat + ((size_t)b * PP + p) * CCH + lane * 4;
#if USE_ASYNC_LDS
      __builtin_amdgcn_global_load_async_to_lds_b128(
          (__attribute__((address_space(1))) v4i_vs*)src,
          (__attribute__((address_space(3))) v4i_vs*)&lds[w * 132 + lane * 4],
          0, 0);
      if (lane == 0) {
        lds[w * 132 + 128] = seg[(size_t)b * PP + p];
        lds[w * 132 + 129] = dep[(size_t)b * PP + p] * (1.0f / 70.0f) - 0.5f;
      }
      wait_async0();
#else
      float4 v = *(const float4*)src;
      lds[w * 132 + lane * 4 + 0] = v.x;
      lds[w * 132 + lane * 4 + 1] = v.y;
      lds[w * 132 + lane * 4 + 2] = v.z;
      lds[w * 132 + lane * 4 + 3] = v.w;
      if (lane == 0) {
        lds[w * 132 + 128] = seg[(size_t)b * PP + p];
        lds[w * 132 + 129] = dep[(size_t)b * PP + p] * (1.0f / 70.0f) - 0.5f;
      }
#endif
    }
    __syncthreads();
    // 8 rows * 130 floats = 1040 contiguous outputs for this group
    for (int u = t; u < 8 * 130; u += 256) {
      int rw = u / 130, c = u % 130;
      float val = 0.0f;
      if (!empty)
        val = (c == 0) ? lds[rw * 132 + 128]
            : (c == 1) ? lds[rw * 132 + 129]
                       : lds[rw * 132 + (c - 2)];
      outf[(size_t)grp * 8 * 130 + u] = val;
    }
    __syncthreads();
  }
}

// ---------------------------------------------------------------------
// Kernel F: sampled xyz = R(ry) * (pool_xyz - center) via fp32 WMMA.
// One wave per ROI; 16 points per V_WMMA_F32_16X16X4_F32:
//   A(16x4) rows = (dx, dy, dz, 0);  B(4x16) cols 0..2 = [c,0,s;0,1,0;-s,0,c]^T
//   D cols 0..2 = (sx, y, sz).
// A layout: lane L -> K=0,1 ; lane L+16 -> K=2,3 (ISA 7.12.2).
// D layout: VGPR r, lanes 0-15 -> M=r ; lanes 16-31 -> M=r+8.
// ---------------------------------------------------------------------
__global__ __launch_bounds__(32) void k_sampled(const float* __restrict__ roi,
                                                const float* __restrict__ xyz,
                                                const int* __restrict__ inds,
                                                const int* __restrict__ idxp,
                                                const int* __restrict__ cnt,
                                                float* __restrict__ out) {
  int rid = blockIdx.x;
  int b = rid >> 6;
  int lane = threadIdx.x;
  int n = lane & 15;
  bool hi = lane >= 16;

  const float* r = roi + (size_t)(b * MM + inds[rid]) * 7;
  float cx = r[0], cy = r[1], cz = r[2], ry = r[6];
  float cc = cosf(ry), ss = sinf(ry);
  bool empty = (cnt[rid] == 0);
  float* o = out + O_SAMPLED + (size_t)rid * NUM_PTS * 3;

#if USE_WMMA_F32
  // B matrix: VGPR0 = rows K=0 (lanes 0-15) / K=2 (lanes 16-31),
  //           VGPR1 = rows K=1 / K=3.
  v2f bmat;
  if (!hi) { bmat.x = (n == 0) ? cc : ((n == 2) ? -ss : 0.f); bmat.y = (n == 1) ? 1.f : 0.f; }
  else     { bmat.x = (n == 0) ? ss : ((n == 2) ?  cc : 0.f); bmat.y = 0.f; }

  for (int it = 0; it < NUM_PTS / 16; ++it) {
    int base = it * 16;
    int q = base + n;                    // lane L and L+16 both hold point base+n
    float px = 0.f, py = 0.f, pz = 0.f;
    if (!empty) {
      int p = idxp[(size_t)rid * NUM_PTS + q];
      const float* s3 = xyz + ((size_t)b * PP + p) * 3;
      px = s3[0]; py = s3[1]; pz = s3[2];
    }
    float dx = px - cx, dyv = py - cy, dz = pz - cz;
    v2f amat;
    amat.x = hi ? dz : dx;               // K=2 : K=0
    amat.y = hi ? 0.f : dyv;             // K=3 : K=1
    v8f acc = {0.f, 0.f, 0.f, 0.f, 0.f, 0.f, 0.f, 0.f};
    v8f d = __builtin_amdgcn_wmma_f32_16x16x4_f32(
        false, amat, false, bmat, (short)0, acc, false, false);
    if (n < 3) {                         // cols 0..2 = sx, y, sz
#pragma unroll
      for (int rr = 0; rr < 8; ++rr) {
        int qq = base + (hi ? 8 + rr : rr);
        o[(size_t)qq * 3 + n] = d[rr];
      }
    }
  }
#else
  for (int it = 0; it < NUM_PTS / 32; ++it) {
    int q = it * 32 + lane;
    float px = 0.f, py = 0.f, pz = 0.f;
    if (!empty) {
      int p = idxp[(size_t)rid * NUM_PTS + q];
      const float* s3 = xyz + ((size_t)b * PP + p) * 3;
      px = s3[0]; py = s3[1]; pz = s3[2];
    }
    float dx = px - cx, dyv = py - cy, dz = pz - cz;
    o[(size_t)q * 3 + 0] = cc * dx + ss * dz;
    o[(size_t)q * 3 + 1] = dyv;
    o[(size_t)q * 3 + 2] = cc * dz - ss * dx;
  }
#endif
}

// ---------------------------------------------------------------------
extern "C" void kernel_launch(void* const* d_in, const int* in_sizes, int n_in,
                              void* d_out, int out_size, void* d_ws, size_t ws_size,
                              hipStream_t stream) {
  (void)in_sizes; (void)n_in; (void)out_size; (void)ws_size;
  const float* roi  = (const float*)d_in[0];   // (4,512,7)
  const float* gt   = (const float*)d_in[1];   // (4,32,8)
  const float* xyz  = (const float*)d_in[2];   // (4,16384,3)
  const float* feat = (const float*)d_in[3];   // (4,16384,128)
  const float* seg  = (const float*)d_in[4];   // (4,16384)
  const float* dep  = (const float*)d_in[5];   // (4,16384)
  float* out = (float*)d_out;

  char* ws = (char*)d_ws;
  float* max_ov = (float*)(ws + 0);            // 4*512 floats
  int*   ga     = (int*)(ws + 8192);           // 4*512 ints
  int*   inds   = (int*)(ws + 16384);          // 256 ints
  int*   cnt    = (int*)(ws + 17408);          // 256 ints
  int*   idxp   = (int*)(ws + 18432);          // 256*512 ints

  k_iou    <<<BB,        512, 0, stream>>>(roi, gt, max_ov, ga);
  k_topk   <<<BB,        512, 0, stream>>>(max_ov, inds);
  k_inside <<<BB * ROI,  512, 0, stream>>>(roi, xyz, inds, idxp, cnt);
  k_labels <<<1,         256, 0, stream>>>(roi, gt, max_ov, ga, inds, cnt, out);
  k_pool   <<<BB * ROI,  256, 0, stream>>>(feat, seg, dep, idxp, cnt, out);
  k_sampled<<<BB * ROI,   32, 0, stream>>>(roi, xyz, inds, idxp, cnt, out);
}